// Node2Vec_54056458387606
// MI455X (gfx1250) — compile-verified
//
#include <hip/hip_runtime.h>

// Node2Vec loss on MI455X (gfx1250, wave32).
// Gather-bandwidth-bound problem; fp32 kept end-to-end via V_WMMA_F32_16X16X4_F32.

typedef __attribute__((ext_vector_type(2))) float v2f;
typedef __attribute__((ext_vector_type(8))) float v8f;

#define N2V_DIM   128   // embedding dim
#define N2V_NPOS  91    // L + 1 + L_NS = 80 + 1 + 10
#define N2V_LWALK 80    // L
#define N2V_NTILE 6     // ceil(91/16) -> 96 padded positions

// One wave (32 lanes) per batch row.
//   A tile: 16 walk positions x 4 K-elems  (lane L: row = node[16t + L%16],
//           V0/V1 hold K = 2*(L/16), 2*(L/16)+1 of each 4-chunk)
//   B tile: start embedding broadcast into all 16 columns -> same float2
//           fetch pattern from the start row.
// D(16x16) f32 accum: lane L holds M = v + 8*(L/16) in acc VGPR v, all 16
// columns identical; lanes with (L%16)==0 extract everything.
__global__ __launch_bounds__(256) void n2v_wmma_kernel(
    const float* __restrict__ X,
    const long long* __restrict__ rw,      // [batch][91] int64 node ids
    float* __restrict__ row_loss,          // [batch]
    int batch)
{
    const int lane = threadIdx.x & 31;
    const int wave = blockIdx.x * (blockDim.x >> 5) + (threadIdx.x >> 5);
    if (wave >= batch) return;             // wave-uniform: EXEC stays all-ones

    const int m    = lane & 15;            // column / M-subindex
    const int half = lane >> 4;            // 0: K={0,1}, 1: K={2,3}
    const int koff = half << 1;

    const long long* rwb = rw + (long long)wave * N2V_NPOS;
    const long long start = rwb[0];
    const float* __restrict__ srow = X + start * (long long)N2V_DIM;

    // Per-tile gathered row base pointers (pad p>=91 with the start row,
    // masked out at extraction time).
    const float* arow[N2V_NTILE];
#pragma unroll
    for (int t = 0; t < N2V_NTILE; ++t) {
        const int p = t * 16 + m;
        const long long idx = (p < N2V_NPOS) ? rwb[p] : start;
        arow[t] = X + idx * (long long)N2V_DIM;
    }

    const v8f vzero = {0.f, 0.f, 0.f, 0.f, 0.f, 0.f, 0.f, 0.f};
    v8f acc[N2V_NTILE];
#pragma unroll
    for (int t = 0; t < N2V_NTILE; ++t) acc[t] = vzero;

    // K loop: 128 / 4 = 32 steps; B reused across the 6 tiles each step.
#pragma unroll 4
    for (int j = 0; j < N2V_DIM / 4; ++j) {
        const int o = 4 * j + koff;
        const v2f b = *(const v2f*)(srow + o);
#pragma unroll
        for (int t = 0; t < N2V_NTILE; ++t) {
            const v2f a = *(const v2f*)(arow[t] + o);
            acc[t] = __builtin_amdgcn_wmma_f32_16x16x4_f32(
                /*neg_a=*/false, a, /*neg_b=*/false, b,
                /*c_mod=*/(short)0, acc[t],
                /*reuse_a=*/false, /*reuse_b=*/false);
        }
    }

    // Extraction: only the N==0 column lanes (lane 0 and lane 16) contribute;
    // between them they cover M = 0..7 (lane 0) and M = 8..15 (lane 16).
    float num = 0.f, esum = 0.f;
    if (m == 0) {
#pragma unroll
        for (int t = 0; t < N2V_NTILE; ++t) {
#pragma unroll
            for (int v = 0; v < 8; ++v) {
                const int p = 16 * t + 8 * half + v;
                const float sc = acc[t][v];
                if (p < N2V_NPOS) {
                    esum += expf(sc);
                    if (p >= 1 && p <= N2V_LWALK) num += sc;
                }
            }
        }
    }
    num  += __shfl_xor(num, 16, 32);
    esum += __shfl_xor(esum, 16, 32);

    if (lane == 0)
        row_loss[wave] = (float)N2V_LWALK * logf(esum) - num;
}

// Deterministic single-block tree reduction (double accumulation) -> mean.
__global__ __launch_bounds__(256) void n2v_reduce_kernel(
    const float* __restrict__ row_loss, float* __restrict__ out, int n)
{
    __shared__ double sm[256];
    double s = 0.0;
    for (int i = threadIdx.x; i < n; i += blockDim.x)
        s += (double)row_loss[i];
    sm[threadIdx.x] = s;
    __syncthreads();
    for (int w = blockDim.x >> 1; w > 0; w >>= 1) {
        if ((int)threadIdx.x < w) sm[threadIdx.x] += sm[threadIdx.x + w];
        __syncthreads();
    }
    if (threadIdx.x == 0) out[0] = (float)(sm[0] / (double)n);
}

extern "C" void kernel_launch(void* const* d_in, const int* in_sizes, int n_in,
                              void* d_out, int out_size, void* d_ws, size_t ws_size,
                              hipStream_t stream) {
    const float* X      = (const float*)d_in[0];       // [1e6][128] fp32
    const long long* rw = (const long long*)d_in[1];   // [batch][91] int64
    // d_in[2] = l (== 80), compile-time constant in this kernel.
    float* out = (float*)d_out;
    float* ws  = (float*)d_ws;                         // batch floats of scratch

    const int batch = in_sizes[1] / N2V_NPOS;          // 8192
    const int waves_per_block = 8;                     // 256 threads = 8 wave32
    const int blocks = (batch + waves_per_block - 1) / waves_per_block;

    n2v_wmma_kernel<<<blocks, waves_per_block * 32, 0, stream>>>(X, rw, ws, batch);
    n2v_reduce_kernel<<<1, 256, 0, stream>>>(ws, out, batch);
}